// DynamicGAT_83665962926553
// MI455X (gfx1250) — compile-verified
//
#include <hip/hip_runtime.h>
#include <hip/hip_bf16.h>
#include <math.h>

typedef __attribute__((ext_vector_type(2))) float v2f;
typedef __attribute__((ext_vector_type(8))) float v8f;
typedef __attribute__((ext_vector_type(4))) unsigned int u32x4;
typedef __attribute__((ext_vector_type(4))) int i32x4;
typedef __attribute__((ext_vector_type(8))) int i32x8;

// Problem constants (from reference)
constexpr int BN  = 8;    // batch
constexpr int NN  = 325;  // nodes
constexpr int FIN = 32;   // input features
constexpr int TT  = 24;   // time steps
constexpr int HH  = 4;    // heads
constexpr int CC  = 32;   // channels per head
constexpr int KK  = 3;    // neighbors per node
constexpr int BT  = BN * TT;          // 192
constexpr int HC  = HH * CC;          // 128
constexpr int NT16 = (NN + 15) / 16;  // 21 row tiles
constexpr int NNP  = NT16 * 16;       // 336: padded rows -> unconditional WMMA stores
constexpr int LDS_STRIDE = 132;       // padded WwT row stride (floats): K-halves 8 banks apart

// ---------------------------------------------------------------------------
// TDM: DMA a flat run of `elems` f32 (16KB here) from global into LDS.
//   D# group0: count=1 | lds_addr | global_addr[56:0] | type=2
//   D# group1: data_size=4B, tensor_dim0=tile_dim0=elems (1D tile)
// ---------------------------------------------------------------------------
__device__ __forceinline__ void tdm_load_f32_to_lds(const float* gsrc, float* lds_dst,
                                                    unsigned int elems) {
  const unsigned long long ga = (unsigned long long)(uintptr_t)gsrc;
  const unsigned int lds_off = (unsigned int)(uintptr_t)(void*)lds_dst; // low 32b = LDS offset
  u32x4 g0;
  g0.x = 1u;                                            // count=1 (valid user descriptor)
  g0.y = lds_off;                                       // lds_addr
  g0.z = (unsigned int)ga;                              // global_addr[31:0]
  g0.w = (unsigned int)((ga >> 32) & 0x1FFFFFFu) | 0x80000000u; // addr[56:32] | type=2
  i32x8 g1;
  g1[0] = 0x20000;                   // workgroup_mask=0, data_size=2 (4 bytes)
  g1[1] = (int)((elems & 0xFFFFu) << 16);               // tensor_dim0[15:0] @ bits 63:48
  g1[2] = (int)(((elems >> 16) & 0xFFFFu)) | (1 << 16); // tensor_dim0[31:16], tensor_dim1=1
  g1[3] = (int)((elems & 0xFFFFu) << 16);               // tile_dim0 @ bits 127:112
  g1[4] = 1;                                            // tile_dim1=1, tile_dim2=0
  g1[5] = (int)elems;                                   // tensor_dim0_stride[31:0]
  g1[6] = 0;
  g1[7] = 0;
  i32x4 z4 = {0, 0, 0, 0};
#if defined(__clang_major__) && (__clang_major__ >= 23)
  i32x8 z8 = {0, 0, 0, 0, 0, 0, 0, 0};
  __builtin_amdgcn_tensor_load_to_lds(g0, g1, z4, z4, z8, 0);
#else
  __builtin_amdgcn_tensor_load_to_lds(g0, g1, z4, z4, 0);
#endif
  __builtin_amdgcn_s_wait_tensorcnt(0);
}

// ---------------------------------------------------------------------------
// Kernel A: Wh[bt, n, hc] = xp[bt, n, :] @ W_w.T + W_b   via V_WMMA_F32_16X16X4_F32
//   One block per bt (21 waves); wave w computes the 16(n) x 128(hc) strip of tile w.
//   W_w staged once per block into LDS by the Tensor Data Mover, then transposed
//   into a bank-padded layout for conflict-free B-fragment ds loads.
//   Wh is padded to NNP rows, so all stores are unconditional (no exec masking).
// A-frag layout (f32 A 16x4): lanes 0-15 -> M=lane, K={0,1}; lanes 16-31 -> M=lane-16, K={2,3}
// C/D layout: VGPR r, lanes 0-15 -> M=r, N=lane; lanes 16-31 -> M=r+8, N=lane-16
// ---------------------------------------------------------------------------
__global__ __launch_bounds__(NT16 * 32) void wh_gemm_wmma(
    const float* __restrict__ x, const float* __restrict__ Ww,
    const float* __restrict__ Wb, float* __restrict__ Wh)
{
  __shared__ float wwRaw[HC * FIN];            // 16 KB TDM landing zone (row-major Ww)
  __shared__ float wwT[FIN * LDS_STRIDE];      // transposed, padded: WwT[f][hc]
  __shared__ float biasS[HC];

  const int bt   = blockIdx.x;
  const int b    = bt / TT;
  const int t    = bt - b * TT;
  const int tid  = threadIdx.x;
  const int wave = tid >> 5;
  const int lane = tid & 31;

  // --- Stage W_w via Tensor Data Mover (wave 0 issues the DMA) ---
  if (wave == 0) {
    tdm_load_f32_to_lds(Ww, wwRaw, HC * FIN);
  }
  if (tid < HC) biasS[tid] = Wb[tid];
  __syncthreads();                              // publish DMA'd data to all waves
  // --- transpose into padded layout: wwT[f*132 + hc] = Ww[hc][f] ---
  for (int e = tid; e < HC * FIN; e += NT16 * 32) {
    const int hc = e >> 5, f = e & 31;
    wwT[f * LDS_STRIDE + hc] = wwRaw[e];
  }
  __syncthreads();

  // --- per-wave 16x128 strip ---
  const int m  = lane & 15;   // A row within tile / output col within tile
  const int kh = lane >> 4;   // K-half selector
  const int n0 = wave * 16;

  const int   n   = n0 + m;
  const int   ncl = (n < NN) ? n : (NN - 1);   // clamp: OOB rows duplicate row 324,
  const float* px = x + ((size_t)(b * NN + ncl) * FIN) * TT + t; // land only in pad rows
  __builtin_prefetch(px, 0, 3);

  v2f afr[8];
#pragma unroll
  for (int k0 = 0; k0 < 8; ++k0) {
    const int f = k0 * 4 + kh * 2;
    afr[k0].x = px[(size_t)f * TT];
    afr[k0].y = px[(size_t)(f + 1) * TT];
  }

#pragma unroll 1
  for (int hcT = 0; hcT < 8; ++hcT) {
    const int hc = hcT * 16 + m;
    v8f acc = {};
#pragma unroll
    for (int k0 = 0; k0 < 8; ++k0) {
      const int f = k0 * 4 + kh * 2;
      v2f bfr;
      bfr.x = wwT[f * LDS_STRIDE + hc];
      bfr.y = wwT[(f + 1) * LDS_STRIDE + hc];
      acc = __builtin_amdgcn_wmma_f32_16x16x4_f32(
          /*neg_a=*/false, afr[k0], /*neg_b=*/false, bfr,
          /*c_mod=*/(short)0, acc, /*reuse_a=*/false, /*reuse_b=*/false);
    }
    const float bias = biasS[hc];
    float* pout = Wh + ((size_t)bt * NNP + (size_t)(n0 + kh * 8)) * HC + hc;
#pragma unroll
    for (int r = 0; r < 8; ++r)
      pout[(size_t)r * HC] = acc[r] + bias;    // unconditional: Wh rows padded to NNP
  }
}

// ---------------------------------------------------------------------------
// Kernel B: f1[bt,n,h] = sum_c Wh[bt,n,h*32+c] * a1[h,c];  f2 likewise with a2.
// ---------------------------------------------------------------------------
__global__ __launch_bounds__(256) void attn_feat_kernel(
    const float* __restrict__ Wh, const float* __restrict__ aw,
    float* __restrict__ f1, float* __restrict__ f2)
{
  const int tid = blockIdx.x * blockDim.x + threadIdx.x;
  if (tid >= BT * NN * HH) return;
  const int h   = tid & (HH - 1);
  const int btn = tid >> 2;
  const int bt  = btn / NN;
  const int n   = btn - bt * NN;
  const float* p  = Wh + ((size_t)bt * NNP + n) * HC + h * CC;
  const float* a1 = aw + h * (2 * CC);
  const float* a2 = a1 + CC;
  float s1 = 0.f, s2 = 0.f;
#pragma unroll
  for (int c = 0; c < CC; ++c) {
    const float v = p[c];
    s1 += v * a1[c];
    s2 += v * a2[c];
  }
  f1[tid] = s1;
  f2[tid] = s2;
}

// ---------------------------------------------------------------------------
// Kernel C: Ssum[bt,hc] = sum_{n<NN} Wh[bt,n,hc]   (pad rows excluded)
// ---------------------------------------------------------------------------
__global__ __launch_bounds__(128) void ssum_kernel(
    const float* __restrict__ Wh, float* __restrict__ Ssum)
{
  const int bt = blockIdx.x;
  const int hc = threadIdx.x;
  const float* p = Wh + (size_t)bt * NNP * HC + hc;
  float s = 0.f;
  for (int n = 0; n < NN; ++n) s += p[(size_t)n * HC];
  Ssum[bt * HC + hc] = s;
}

// ---------------------------------------------------------------------------
// Kernel D: w = _constrain(iw, base, adj); extract the K=3 neighbors per row.
// ---------------------------------------------------------------------------
__global__ __launch_bounds__(128) void constrain_nbrs(
    const float* __restrict__ iw, const float* __restrict__ base,
    const float* __restrict__ adj, int* __restrict__ nidx, float* __restrict__ nw)
{
  const int i   = blockIdx.x;
  const int tid = threadIdx.x;
  __shared__ float red[128];
  __shared__ int   cnt;
  __shared__ int   sj[8];
  __shared__ float sw[8];
  if (tid == 0) cnt = 0;

  float part = 0.f;
  for (int j = tid; j < NN; j += 128) {
    const float bb = base[(size_t)i * NN + j];
    float v = fminf(fmaxf(iw[(size_t)i * NN + j], 0.5f * bb), 1.5f * bb);
    v = fmaxf(v, 0.f);
    part += v * adj[(size_t)i * NN + j];
  }
  red[tid] = part;
  __syncthreads();
  for (int s = 64; s > 0; s >>= 1) {
    if (tid < s) red[tid] += red[tid + s];
    __syncthreads();
  }
  float ssum = red[0];
  ssum += (ssum == 0.f) ? 1e-6f : 0.f;

  for (int j = tid; j < NN; j += 128) {
    const float a = adj[(size_t)i * NN + j];
    if (a != 0.f) {
      const float bb = base[(size_t)i * NN + j];
      float v = fminf(fmaxf(iw[(size_t)i * NN + j], 0.5f * bb), 1.5f * bb);
      v = fmaxf(v, 0.f);
      const int pos = atomicAdd(&cnt, 1);
      if (pos < 8) { sj[pos] = j; sw[pos] = v * a / ssum; }
    }
  }
  __syncthreads();
  if (tid < KK) {
    nidx[i * KK + tid] = sj[tid];
    nw[i * KK + tid]   = sw[tid];
  }
}

// ---------------------------------------------------------------------------
// Kernel E: closed-form K-sparse softmax + output + final transpose.
//   Off-mask scores are exactly 0 (322 of them per row):
//   out[b,i,hc,t] = p_off*Ssum + sum_k (p_k - p_off)*Wh[j_k]
// ---------------------------------------------------------------------------
__global__ __launch_bounds__(128) void attn_out_kernel(
    const float* __restrict__ Wh, const float* __restrict__ f1,
    const float* __restrict__ f2, const float* __restrict__ Ssum,
    const int* __restrict__ nidx, const float* __restrict__ nw,
    float* __restrict__ out)
{
  const int blk = blockIdx.x;         // bt*NN + i
  const int bt  = blk / NN;
  const int i   = blk - bt * NN;
  const int b   = bt / TT;
  const int t   = bt - b * TT;
  const int hc  = threadIdx.x;
  const int h   = hc >> 5;

  const int   j0 = nidx[i * KK + 0], j1 = nidx[i * KK + 1], j2 = nidx[i * KK + 2];
  const float w0 = nw[i * KK + 0],   w1 = nw[i * KK + 1],   w2 = nw[i * KK + 2];

  const float fi = f1[((size_t)bt * NN + i) * HH + h];
  float g0 = fi + f2[((size_t)bt * NN + j0) * HH + h];
  float g1 = fi + f2[((size_t)bt * NN + j1) * HH + h];
  float g2 = fi + f2[((size_t)bt * NN + j2) * HH + h];
  g0 = (g0 > 0.f) ? g0 : 0.2f * g0;   // leaky_relu 0.2
  g1 = (g1 > 0.f) ? g1 : 0.2f * g1;
  g2 = (g2 > 0.f) ? g2 : 0.2f * g2;
  const float s0 = g0 + w0, s1 = g1 + w1, s2 = g2 + w2;

  const float mx = fmaxf(fmaxf(s0, s1), fmaxf(s2, 0.f));
  const float e0 = __expf(s0 - mx), e1 = __expf(s1 - mx), e2 = __expf(s2 - mx);
  const float eo = __expf(-mx);
  const float Z   = (float)(NN - KK) * eo + e0 + e1 + e2;
  const float inv = 1.f / Z;
  const float po = eo * inv;
  const float p0 = e0 * inv - po, p1 = e1 * inv - po, p2 = e2 * inv - po;

  const size_t basew = (size_t)bt * NNP * HC;
  const float v = po * Ssum[bt * HC + hc]
                + p0 * Wh[basew + (size_t)j0 * HC + hc]
                + p1 * Wh[basew + (size_t)j1 * HC + hc]
                + p2 * Wh[basew + (size_t)j2 * HC + hc];

  out[(((size_t)b * NN + i) * HC + hc) * TT + t] = v;
}

// ---------------------------------------------------------------------------
extern "C" void kernel_launch(void* const* d_in, const int* in_sizes, int n_in,
                              void* d_out, int out_size, void* d_ws, size_t ws_size,
                              hipStream_t stream)
{
  (void)in_sizes; (void)n_in; (void)out_size; (void)ws_size;
  const float* x   = (const float*)d_in[0];  // (B,N,F_IN,T)
  const float* Ww  = (const float*)d_in[1];  // (HC, F_IN)
  const float* Wb  = (const float*)d_in[2];  // (HC,)
  const float* aw  = (const float*)d_in[3];  // (H, 2C)
  const float* iw  = (const float*)d_in[4];  // (N,N)
  const float* iwb = (const float*)d_in[5];  // (N,N)
  const float* adj = (const float*)d_in[6];  // (N,N)
  float* out = (float*)d_out;

  char* ws = (char*)d_ws;
  size_t off = 0;
  auto take = [&](size_t bytes) -> char* {
    char* p = ws + off;
    off = (off + bytes + 255) & ~(size_t)255;
    return p;
  };
  float* Wh   = (float*)take((size_t)BT * NNP * HC * sizeof(float));  // ~33.0 MB (padded)
  float* f1   = (float*)take((size_t)BT * NN * HH * sizeof(float));
  float* f2   = (float*)take((size_t)BT * NN * HH * sizeof(float));
  float* Ssum = (float*)take((size_t)BT * HC * sizeof(float));
  int*   nidx = (int*)  take((size_t)NN * KK * sizeof(int));
  float* nwgt = (float*)take((size_t)NN * KK * sizeof(float));

  wh_gemm_wmma<<<BT, NT16 * 32, 0, stream>>>(x, Ww, Wb, Wh);
  constrain_nbrs<<<NN, 128, 0, stream>>>(iw, iwb, adj, nidx, nwgt);
  attn_feat_kernel<<<(BT * NN * HH + 255) / 256, 256, 0, stream>>>(Wh, aw, f1, f2);
  ssum_kernel<<<BT, 128, 0, stream>>>(Wh, Ssum);
  attn_out_kernel<<<BT * NN, 128, 0, stream>>>(Wh, f1, f2, Ssum, nidx, nwgt, out);
}